// RM_Network_88089779241066
// MI455X (gfx1250) — compile-verified
//
#include <hip/hip_runtime.h>
#include <hip/hip_bf16.h>

typedef __attribute__((ext_vector_type(2))) float v2f;
typedef __attribute__((ext_vector_type(8))) float v8f;

#define CCH   64          // channels (independent Phi nets)
#define BATCH 2048
#define HID   8
#define FEATN 128
#define HW    361         // 19*19 input spatial
#define OHW   289         // 17*17 conv-out spatial
#define EPSV  1e-5f
#define FSTR  132         // padded feature row stride (bank-conflict-free b64)

#define AS1 __attribute__((address_space(1)))
#define AS3 __attribute__((address_space(3)))

#if __has_builtin(__builtin_amdgcn_global_load_async_to_lds_b32) && \
    __has_builtin(__builtin_amdgcn_s_wait_asynccnt)
#define USE_ASYNC_LDS 1
#else
#define USE_ASYNC_LDS 0
#endif

// Stage one batch row of x (both input channels, 2x19x19) into LDS with
// padded rows [2][19][20]. Uses the CDNA5 async global->LDS path when the
// toolchain exposes it (ASYNCcnt-tracked, bypasses VGPRs).
__device__ __forceinline__
void stage_x(const float* __restrict__ p1, const float* __restrict__ p2,
             float* xs, int t)
{
#if USE_ASYNC_LDS
    for (int i = t; i < HW; i += 128) {
        const int r = i / 19, cc = i - r * 19;
        __builtin_amdgcn_global_load_async_to_lds_b32(
            (AS1 int*)(uintptr_t)(p1 + i),
            (AS3 int*)(unsigned)(uintptr_t)&xs[r * 20 + cc], 0, 0);
        __builtin_amdgcn_global_load_async_to_lds_b32(
            (AS1 int*)(uintptr_t)(p2 + i),
            (AS3 int*)(unsigned)(uintptr_t)&xs[380 + r * 20 + cc], 0, 0);
    }
    __builtin_amdgcn_s_wait_asynccnt(0);   // LDS writes visible before barrier
#else
    for (int i = t; i < HW; i += 128) {
        const int r = i / 19, cc = i - r * 19;
        xs[r * 20 + cc]       = p1[i];
        xs[380 + r * 20 + cc] = p2[i];
    }
#endif
}

// ---------------------------------------------------------------------------
// Pass 1: conv + bias over full 17x17, accumulate per-(c,och) sum / sumsq
// for training-mode BatchNorm statistics. grid = (BATCH/16, CCH), block = 128.
// ---------------------------------------------------------------------------
__global__ __launch_bounds__(128)
void k_stats(const float* __restrict__ x1, const float* __restrict__ x2,
             const float* __restrict__ cw, const float* __restrict__ cb,
             float* __restrict__ stats)
{
    __shared__ float xs[2 * 19 * 20];   // [ich][19][20] padded rows
    __shared__ float wsh[36];           // [och][ich][3][3]
    __shared__ float cbs[2];
    const int t  = threadIdx.x;
    const int c  = blockIdx.y;
    const int b0 = blockIdx.x * 16;

    if (t < 36) wsh[t] = cw[c * 36 + t];
    if (t < 2)  cbs[t] = cb[c * 2 + t];

    float s0 = 0.f, q0 = 0.f, s1 = 0.f, q1 = 0.f;

    for (int ib = 0; ib < 16; ++ib) {
        const int b = b0 + ib;
        const float* p1 = x1 + (b * CCH + c) * HW;   // ich0 = x1[:,c]
        const float* p2 = x2 + (b * CCH + c) * HW;   // ich1 = x2[:,c]
        __syncthreads();
        stage_x(p1, p2, xs, t);
        if (ib + 1 < 16) {   // hint next batch row toward the caches
            __builtin_prefetch(p1 + CCH * HW, 0, 0);
            __builtin_prefetch(p2 + CCH * HW, 0, 0);
        }
        __syncthreads();
        for (int idx = t; idx < 2 * OHW; idx += 128) {
            const int och = (idx >= OHW);
            const int rem = idx - och * OHW;
            const int oh = rem / 17, ow = rem - oh * 17;
            float acc = cbs[och];
            #pragma unroll
            for (int ich = 0; ich < 2; ++ich)
                #pragma unroll
                for (int kh = 0; kh < 3; ++kh)
                    #pragma unroll
                    for (int kw = 0; kw < 3; ++kw)
                        acc += wsh[(och * 2 + ich) * 9 + kh * 3 + kw] *
                               xs[ich * 380 + (oh + kh) * 20 + (ow + kw)];
            const float a0 = (och == 0) ? acc : 0.f;
            const float a1 = (och == 1) ? acc : 0.f;
            s0 += a0; q0 += a0 * a0;
            s1 += a1; q1 += a1 * a1;
        }
    }
    #pragma unroll
    for (int m = 1; m < 32; m <<= 1) {
        s0 += __shfl_xor(s0, m, 32);
        q0 += __shfl_xor(q0, m, 32);
        s1 += __shfl_xor(s1, m, 32);
        q1 += __shfl_xor(q1, m, 32);
    }
    if ((t & 31) == 0) {
        atomicAdd(&stats[c * 4 + 0], s0);
        atomicAdd(&stats[c * 4 + 1], q0);
        atomicAdd(&stats[c * 4 + 2], s1);
        atomicAdd(&stats[c * 4 + 3], q1);
    }
}

// ---------------------------------------------------------------------------
// Fold BN stats + gamma/beta into per-(c,och) affine: y' = a*conv + d
// ---------------------------------------------------------------------------
__global__ void k_bn(const float* __restrict__ stats,
                     const float* __restrict__ g, const float* __restrict__ bb,
                     float* __restrict__ affine)
{
    const int i = threadIdx.x;              // i = c*2 + och, 0..127
    if (i < 128) {
        const float n    = (float)BATCH * (float)OHW;
        const float mean = stats[i * 2 + 0] / n;
        const float var  = stats[i * 2 + 1] / n - mean * mean;
        const float a    = g[i] * rsqrtf(var + EPSV);
        affine[i * 2 + 0] = a;
        affine[i * 2 + 1] = bb[i] - mean * a;
    }
}

// ---------------------------------------------------------------------------
// Pass 2: recompute conv on the pooled 16x16 region, affine+LeakyReLU+maxpool
// -> feat[16][128] in LDS; wave 0 does fc1 via V_WMMA_F32_16X16X4_F32
// (K=128 as 32 WMMA steps, N padded 8->16 with zeroed B rows so the fragment
// load is unconditional -- no EXEC juggling in the WMMA loop), then fc2 and
// the channel-mix contribution via atomicAdd into logits.
// grid = (BATCH/16, CCH), block = 128.
// ---------------------------------------------------------------------------
__global__ __launch_bounds__(128)
void k_main(const float* __restrict__ x1, const float* __restrict__ x2,
            const float* __restrict__ cw, const float* __restrict__ cb,
            const float* __restrict__ w1, const float* __restrict__ b1,
            const float* __restrict__ w2, const float* __restrict__ b2,
            const float* __restrict__ wout,
            const float* __restrict__ affine,
            float* __restrict__ logits)
{
    __shared__ float xs[2 * 19 * 20];
    __shared__ float featL[16 * FSTR];       // [batch-row][128] padded
    __shared__ float w1s[16 * FSTR];         // fc1_w[c] padded; rows 8..15 = 0
    __shared__ float wsh[36];
    __shared__ float prm[6];                 // cb0,cb1, a0,a1, d0,d1
    const int t  = threadIdx.x;
    const int c  = blockIdx.y;
    const int b0 = blockIdx.x * 16;

    for (int i = t; i < HID * FEATN; i += 128) {
        const int n = i >> 7, k = i & 127;
        w1s[n * FSTR + k] = w1[c * HID * FEATN + i];
    }
    for (int i = t; i < (16 - HID) * FSTR; i += 128)   // zero N-pad rows
        w1s[HID * FSTR + i] = 0.f;
    if (t < 36) wsh[t] = cw[c * 36 + t];
    if (t < 2) {
        prm[t]     = cb[c * 2 + t];
        prm[2 + t] = affine[(c * 2 + t) * 2 + 0];
        prm[4 + t] = affine[(c * 2 + t) * 2 + 1];
    }

    const int och = t >> 6, ph = (t >> 3) & 7, pw = t & 7;

    for (int ib = 0; ib < 16; ++ib) {
        const int b = b0 + ib;
        const float* p1 = x1 + (b * CCH + c) * HW;
        const float* p2 = x2 + (b * CCH + c) * HW;
        __syncthreads();
        stage_x(p1, p2, xs, t);
        if (ib + 1 < 16) {
            __builtin_prefetch(p1 + CCH * HW, 0, 0);
            __builtin_prefetch(p2 + CCH * HW, 0, 0);
        }
        __syncthreads();
        const float cb0 = prm[och], av = prm[2 + och], dv = prm[4 + och];
        float mval = -3.0e38f;
        #pragma unroll
        for (int dy = 0; dy < 2; ++dy)
            #pragma unroll
            for (int dx = 0; dx < 2; ++dx) {
                const int oh = 2 * ph + dy, ow = 2 * pw + dx;
                float acc = cb0;
                #pragma unroll
                for (int ich = 0; ich < 2; ++ich)
                    #pragma unroll
                    for (int kh = 0; kh < 3; ++kh)
                        #pragma unroll
                        for (int kw = 0; kw < 3; ++kw)
                            acc += wsh[(och * 2 + ich) * 9 + kh * 3 + kw] *
                                   xs[ich * 380 + (oh + kh) * 20 + (ow + kw)];
                float y = av * acc + dv;                 // folded BatchNorm
                y = (y > 0.f) ? y : 0.01f * y;           // LeakyReLU
                mval = fmaxf(mval, y);                   // maxpool 2x2
            }
        featL[ib * FSTR + t] = mval;   // t == och*64 + ph*8 + pw == flatten idx
        __syncthreads();
    }
    __syncthreads();

    if (t < 32) {                       // wave 0: EXEC all ones, wave32 WMMA
        const int n = t & 15, g = t >> 4;     // n = N col / batch row, g = half
        v8f acc = {};
        #pragma unroll
        for (int kb = 0; kb < 32; ++kb) {
            const int k0 = kb * 4 + 2 * g;    // A/B lane-half K offset
            v2f afrag = *(const v2f*)&featL[n * FSTR + k0];   // A[m=n][k0,k0+1]
            v2f bfrag = *(const v2f*)&w1s[n * FSTR + k0];     // rows>=8 are 0
            acc = __builtin_amdgcn_wmma_f32_16x16x4_f32(
                false, afrag, false, bfrag, (short)0, acc, false, false);
        }
        // acc[v] = fc1out[batch = v + 8*g][hidden = n] (exact 0 for n >= 8)
        const float b1n = (n < HID) ? b1[c * HID + n] : 0.f;
        const float w2n = (n < HID) ? w2[c * HID + n] : 0.f;
        const float b2c = b2[c];
        const float wo  = wout[c];
        float tv[8];
        #pragma unroll
        for (int v = 0; v < 8; ++v) tv[v] = (acc[v] + b1n) * w2n;
        #pragma unroll
        for (int m = 1; m < 16; m <<= 1) {    // reduce over hidden within half
            #pragma unroll
            for (int v = 0; v < 8; ++v) tv[v] += __shfl_xor(tv[v], m, 32);
        }
        #pragma unroll
        for (int v = 0; v < 8; ++v)
            if (n == v)
                atomicAdd(&logits[b0 + v + 8 * g], (tv[v] + b2c) * wo);
    }
}

// ---------------------------------------------------------------------------
// Final: sigmoid(logits + b_out)
// ---------------------------------------------------------------------------
__global__ void k_out(const float* __restrict__ logits,
                      const float* __restrict__ bout, float* __restrict__ out)
{
    const int i = blockIdx.x * 256 + threadIdx.x;
    if (i < BATCH) {
        const float z = logits[i] + bout[0];
        out[i] = 1.f / (1.f + expf(-z));
    }
}

// ---------------------------------------------------------------------------
extern "C" void kernel_launch(void* const* d_in, const int* in_sizes, int n_in,
                              void* d_out, int out_size, void* d_ws, size_t ws_size,
                              hipStream_t stream)
{
    const float* x1 = (const float*)d_in[0];
    const float* x2 = (const float*)d_in[1];
    const float* cw = (const float*)d_in[2];
    const float* cb = (const float*)d_in[3];
    const float* bg = (const float*)d_in[4];
    const float* bb = (const float*)d_in[5];
    const float* w1 = (const float*)d_in[6];
    const float* b1 = (const float*)d_in[7];
    const float* w2 = (const float*)d_in[8];
    const float* b2 = (const float*)d_in[9];
    const float* wo = (const float*)d_in[10];
    const float* bo = (const float*)d_in[11];

    float* ws     = (float*)d_ws;
    float* stats  = ws;               // 256 floats  [C][och][sum,sumsq]
    float* logits = ws + 256;         // 2048 floats
    float* affine = ws + 256 + 2048;  // 256 floats  [C][och][a,d]

    // zero stats + logits every call (graph-capture-safe memset node)
    (void)hipMemsetAsync(ws, 0, (256 + 2048) * sizeof(float), stream);

    const dim3 blk(128);
    const dim3 grd(BATCH / 16, CCH);
    hipLaunchKernelGGL(k_stats, grd, blk, 0, stream, x1, x2, cw, cb, stats);
    hipLaunchKernelGGL(k_bn, dim3(1), dim3(128), 0, stream, stats, bg, bb, affine);
    hipLaunchKernelGGL(k_main, grd, blk, 0, stream,
                       x1, x2, cw, cb, w1, b1, w2, b2, wo, affine, logits);
    hipLaunchKernelGGL(k_out, dim3(BATCH / 256), dim3(256), 0, stream,
                       logits, bo, (float*)d_out);
}